// RadonTransform_28595892257239
// MI455X (gfx1250) — compile-verified
//
#include <hip/hip_runtime.h>
#include <math.h>

#define IMG       256
#define ANG       360
#define DET       384
#define ROWSTRIDE 392   // 392 mod 64 = 8 -> stagger LDS banks across wave rows
#define PIXELS    (IMG * IMG)

typedef __attribute__((ext_vector_type(2))) float v2f;
typedef __attribute__((ext_vector_type(4))) float v4f;
typedef __attribute__((ext_vector_type(8))) float v8f;

__global__ __launch_bounds__(256) void radon_fwd_kernel(const float* __restrict__ x,
                                                        float* __restrict__ out) {
    __shared__ float acc[8 * ROWSTRIDE];   // one 384-bin row per wave (padded)

    const int tid  = threadIdx.x;
    const int wave = tid >> 5;
    const int lane = tid & 31;
    const int blk  = blockIdx.x;
    const int b    = blk / ANG;            // batch index
    const int a    = blk - b * ANG;        // angle index

    // zero all accumulator rows (incl. padding)
    for (int i = tid; i < 8 * ROWSTRIDE; i += 256) acc[i] = 0.0f;
    __syncthreads();

    // angle = a * pi/360, computed as f32 like the reference
    const float ang = (float)((double)a * (3.14159265358979323846 / 360.0));
    float sn, cs;
    sincosf(ang, &sn, &cs);

    const float* img    = x + (size_t)b * PIXELS;
    float*       accrow = &acc[wave * ROWSTRIDE];

    // pixel-driven splat: 4 consecutive pixels per iteration (one b128 load).
    // Rows are 256 pixels, quads never straddle a row.
    for (int p = tid * 4; p < PIXELS; p += 256 * 4) {
        v4f pix = *(const v4f*)(img + p);                    // global_load_b128
        float px0  = (float)(p & (IMG - 1)) - 127.5f;        // X of first pixel
        float py   = 127.5f - (float)(p >> 8);               // Y (same for all 4)
        float ysum = fmaf(py, sn, 191.5f);                   // y*sin + center
#pragma unroll
        for (int k = 0; k < 4; ++k) {
            float v    = (pix[k] + 1.0f) * 0.5f;             // [-1,1] -> [0,1]
            float dpos = fmaf(px0 + (float)k, cs, ysum);
            float d0f  = floorf(dpos);
            float fr   = dpos - d0f;
            int   d0   = (int)d0f;
            d0 = max(0, min(d0, DET - 2));                   // defensive clamp (never hit)
            atomicAdd(&accrow[d0],     v * (1.0f - fr));     // ds_add_f32
            atomicAdd(&accrow[d0 + 1], v * fr);              // ds_add_f32
        }
    }
    __syncthreads();

    // fold 8 rows -> 4 rows with plain adds
    for (int i = tid; i < 4 * ROWSTRIDE; i += 256) acc[i] += acc[i + 4 * ROWSTRIDE];
    __syncthreads();

    // Column-sum the 4 remaining rows with V_WMMA_F32_16X16X4_F32:
    // D = A(16x4, all ones) * B(4x16 chunk of rows) + 0 -> every row of D = column sums.
    // Each wave reduces 48 detector bins (3 chunks of 16). EXEC is full for all waves.
    float* obase  = out + ((size_t)b * ANG + a) * DET;
    const int nloc  = lane & 15;   // N index within chunk (B/C/D stripe rows across lanes)
    const int khalf = lane >> 4;   // which K pair this half-wave supplies

    v2f A = {1.0f, 1.0f};
    for (int j = 0; j < 3; ++j) {
        int n0 = wave * 48 + j * 16;
        int n  = n0 + nloc;
        v2f B;
        B.x = acc[(khalf)     * ROWSTRIDE + n];
        B.y = acc[(khalf + 2) * ROWSTRIDE + n];
        v8f C = {};
        v8f D = __builtin_amdgcn_wmma_f32_16x16x4_f32(
            /*neg_a=*/false, A, /*neg_b=*/false, B,
            /*c_mod=*/(short)0, C, /*reuse_a=*/false, /*reuse_b=*/false);
        // All D rows equal the column sums; lanes L and L+16 store the same value
        // to the same address (benign duplicate store, full coverage of 384 bins).
        obase[n] = D[0];
    }
}

extern "C" void kernel_launch(void* const* d_in, const int* in_sizes, int n_in,
                              void* d_out, int out_size, void* d_ws, size_t ws_size,
                              hipStream_t stream) {
    const float* x   = (const float*)d_in[0];
    float*       out = (float*)d_out;
    const int batches = in_sizes[0] / PIXELS;      // 4*1 images
    dim3 grid(batches * ANG);                      // one WG per (batch, angle)
    dim3 block(256);                               // 8 wave32 waves
    radon_fwd_kernel<<<grid, block, 0, stream>>>(x, out);
}